// Encoder_14018773254741
// MI455X (gfx1250) — compile-verified
//
#include <hip/hip_runtime.h>

typedef __attribute__((ext_vector_type(2))) float v2f;
typedef __attribute__((ext_vector_type(8))) float v8f;

#define N_NODES 40000
#define N_EDGES 640000
#define DIM 128
#define DECD 64

__device__ __forceinline__ unsigned pcg(unsigned v) {
    v = v * 747796405u + 2891336453u;
    unsigned w = ((v >> ((v >> 28u) + 4u)) ^ v) * 277803737u;
    return (w >> 22u) ^ w;
}

__global__ void zero_f32(float* __restrict__ p, int n) {
    int i = blockIdx.x * blockDim.x + threadIdx.x;
    int stride = gridDim.x * blockDim.x;
    for (; i < n; i += stride) p[i] = 0.0f;
}

__global__ void deg_count(const int* __restrict__ dst, float* __restrict__ deg, int nE) {
    int i = blockIdx.x * blockDim.x + threadIdx.x;
    if (i < nE) atomicAdd(&deg[dst[i]], 1.0f);
}

__global__ void inv_degp1(float* __restrict__ deg, int n) {
    int i = blockIdx.x * blockDim.x + threadIdx.x;
    if (i < n) deg[i] = 1.0f / (deg[i] + 1.0f);
}

__global__ void dropout_x(const float* __restrict__ x, float* __restrict__ h, int n, unsigned salt) {
    int i = blockIdx.x * blockDim.x + threadIdx.x;
    if (i < n) {
        unsigned r = pcg((unsigned)i ^ salt);
        h[i] = (r & 1u) ? x[i] * 2.0f : 0.0f;
    }
}

// One wave per edge: 32 lanes x float4 = 128 features, 4 f32 atomic adds per lane.
__global__ void scatter_add(const float* __restrict__ h, const int* __restrict__ src,
                            const int* __restrict__ dst, float* __restrict__ neigh, int nE) {
    int gtid = blockIdx.x * blockDim.x + threadIdx.x;
    int e    = gtid >> 5;
    int lane = gtid & 31;
    if (e >= nE) return;
    int s = src[e], d = dst[e];
    float4 v = ((const float4*)(h + (size_t)s * DIM))[lane];
    float* nb = neigh + (size_t)d * DIM + lane * 4;
    atomicAdd(nb + 0, v.x);
    atomicAdd(nb + 1, v.y);
    atomicAdd(nb + 2, v.z);
    atomicAdd(nb + 3, v.w);
}

// Fused SAGE-GCN dense layer: out = f( [(neigh + h) * invdeg] @ W^T + bias )
// One wave computes one 16x16 output tile via 32 chained V_WMMA_F32_16X16X4_F32.
// DO_AGG/DO_RELU/DO_DROP are compile-time so the K-loop is branch-free straight
// line code (global_load_b64 pairs -> v_pk_add/v_pk_mul -> v_wmma).
// All branches are wave-uniform -> EXEC all ones at every WMMA (ISA requirement).
template <int DO_AGG, int DO_RELU, int DO_DROP>
__global__ void wmma_gemm(const float* __restrict__ neigh, const float* __restrict__ hself,
                          const float* __restrict__ invdeg, const float* __restrict__ W,
                          const float* __restrict__ bias, float* __restrict__ out,
                          int nRowTiles, int nColTiles, int outStride, unsigned dropSalt) {
    int lane = threadIdx.x & 31;
    int wave = blockIdx.x * (blockDim.x >> 5) + (threadIdx.x >> 5);
    if (wave >= nRowTiles * nColTiles) return;   // wave-uniform exit
    int rowTile = wave / nColTiles;
    int colTile = wave % nColTiles;
    int m    = lane & 15;
    int half = lane >> 4;                        // K pair {0,1} vs {2,3}
    int row = (rowTile << 4) + m;
    int col = (colTile << 4) + m;

    const float2* hrow = (const float2*)(hself + (size_t)row * DIM);
    const float2* nrow = (const float2*)(neigh + (size_t)row * DIM);
    const float2* wrow = (const float2*)(W     + (size_t)col * DIM);
    float idg = DO_AGG ? invdeg[row] : 1.0f;

    v8f c = {};
#pragma unroll
    for (int k = 0; k < DIM; k += 4) {
        int p = (k >> 1) + half;                 // float2 index of (k + 2*half)
        float2 hv = hrow[p];
        if (DO_AGG) {
            float2 nv = nrow[p];
            hv.x = (hv.x + nv.x) * idg;
            hv.y = (hv.y + nv.y) * idg;
        }
        float2 wv = wrow[p];
        v2f a; a.x = hv.x; a.y = hv.y;
        v2f b; b.x = wv.x; b.y = wv.y;
        c = __builtin_amdgcn_wmma_f32_16x16x4_f32(false, a, false, b, (short)0, c,
                                                  false, false);
    }

    float bv = bias[col];
#pragma unroll
    for (int j = 0; j < 8; ++j) {
        int orow = (rowTile << 4) + j + half * 8;
        float v = c[j] + bv;
        if (DO_RELU) v = fmaxf(v, 0.0f);
        if (DO_DROP) {
            unsigned r = pcg((unsigned)(orow * outStride + col) ^ dropSalt);
            v = (r & 1u) ? v * 2.0f : 0.0f;
        }
        out[(size_t)orow * outStride + col] = v;
    }
}

// One wave per node: log-softmax over 64 logits (2 per lane) + soft-target CE.
// h_b logits for node i are just logitsA[shuffled[i]] (same GEMM, permuted rows).
__global__ void decoder_loss(const float* __restrict__ logits, const int* __restrict__ shuf,
                             float* __restrict__ lossOut, int nNodes) {
    int gtid = blockIdx.x * blockDim.x + threadIdx.x;
    int node = gtid >> 5;
    int lane = gtid & 31;
    if (node >= nNodes) return;
    const float* la = logits + (size_t)node * DECD;
    float x0 = la[lane], x1 = la[lane + 32];
    float mx = fmaxf(x0, x1);
#pragma unroll
    for (int off = 16; off; off >>= 1) mx = fmaxf(mx, __shfl_xor(mx, off, 32));
    float e = expf(x0 - mx) + expf(x1 - mx);
#pragma unroll
    for (int off = 16; off; off >>= 1) e += __shfl_xor(e, off, 32);
    float lse = logf(e) + mx;
    const float* lb = logits + (size_t)shuf[node] * DECD;
    float loss = -(lb[lane] * (x0 - lse) + lb[lane + 32] * (x1 - lse));
#pragma unroll
    for (int off = 16; off; off >>= 1) loss += __shfl_xor(loss, off, 32);
    if (lane == 0) atomicAdd(lossOut, loss * (1.0f / (float)nNodes));
}

extern "C" void kernel_launch(void* const* d_in, const int* in_sizes, int n_in,
                              void* d_out, int out_size, void* d_ws, size_t ws_size,
                              hipStream_t stream) {
    const float* x   = (const float*)d_in[0];
    const int*   src = (const int*)d_in[1];
    const int*   dst = (const int*)d_in[2];
    const int*   shf = (const int*)d_in[3];
    const float* W1  = (const float*)d_in[4];
    const float* b1  = (const float*)d_in[5];
    const float* W2  = (const float*)d_in[6];
    const float* b2  = (const float*)d_in[7];
    const float* Wd  = (const float*)d_in[8];
    const float* bd  = (const float*)d_in[9];

    float* h2   = (float*)d_out;                         // [N,128] output h
    float* loss = (float*)d_out + (size_t)N_NODES * DIM; // scalar dec

    float* ws     = (float*)d_ws;
    float* h0     = ws;  ws += (size_t)N_NODES * DIM;    // dropout(x)
    float* neigh  = ws;  ws += (size_t)N_NODES * DIM;    // segment sums (reused)
    float* h1     = ws;  ws += (size_t)N_NODES * DIM;    // layer1 out (relu+drop)
    float* logitA = ws;  ws += (size_t)N_NODES * DECD;   // decoder logits
    float* invdg  = ws;                                  // 1/(deg+1)

    const unsigned SALT1 = 0x9E3779B9u, SALT2 = 0x85EBCA6Bu;
    const int nElem = N_NODES * DIM;
    const int rowTiles = N_NODES / 16;                   // 2500

    // degrees (shared by both layers)
    zero_f32<<<512, 256, 0, stream>>>(invdg, N_NODES);
    deg_count<<<(N_EDGES + 255) / 256, 256, 0, stream>>>(dst, invdg, N_EDGES);
    inv_degp1<<<(N_NODES + 255) / 256, 256, 0, stream>>>(invdg, N_NODES);

    // h0 = dropout(x)
    dropout_x<<<(nElem + 255) / 256, 256, 0, stream>>>(x, h0, nElem, SALT1);

    // layer 1
    zero_f32<<<4096, 256, 0, stream>>>(neigh, nElem);
    scatter_add<<<(N_EDGES * 32) / 256, 256, 0, stream>>>(h0, src, dst, neigh, N_EDGES);
    wmma_gemm<1, 1, 1><<<(rowTiles * 8 + 7) / 8, 256, 0, stream>>>(
        neigh, h0, invdg, W1, b1, h1, rowTiles, 8, DIM, SALT2);

    // layer 2 -> h2 (first part of d_out)
    zero_f32<<<4096, 256, 0, stream>>>(neigh, nElem);
    scatter_add<<<(N_EDGES * 32) / 256, 256, 0, stream>>>(h1, src, dst, neigh, N_EDGES);
    wmma_gemm<1, 0, 0><<<(rowTiles * 8 + 7) / 8, 256, 0, stream>>>(
        neigh, h1, invdg, W2, b2, h2, rowTiles, 8, DIM, 0u);

    // decoder logits (no aggregation): logitA = h2 @ Wd^T + bd
    wmma_gemm<0, 0, 0><<<(rowTiles * 4 + 7) / 8, 256, 0, stream>>>(
        neigh, h2, invdg, Wd, bd, logitA, rowTiles, 4, DECD, 0u);

    // soft-target CE, mean over nodes
    zero_f32<<<1, 32, 0, stream>>>(loss, 1);
    decoder_loss<<<(N_NODES * 32) / 256, 256, 0, stream>>>(logitA, shf, loss, N_NODES);
}